// GMM_73658689126814
// MI455X (gfx1250) — compile-verified
//
#include <hip/hip_runtime.h>
#include <hip/hip_bf16.h>

// ---------------------------------------------------------------------------
// GMMConv GNN forward for MI455X (gfx1250), wave32.
// Node GEMMs use V_WMMA_F32_16X16X4_F32 (full fp32 precision; GEMMs are a
// negligible fraction of runtime, so no reason to drop precision).
// Edge scatter is the dominant cost; hg/agg (12.8 MB each) are L2-resident.
// ---------------------------------------------------------------------------

typedef __attribute__((ext_vector_type(2))) float v2f;
typedef __attribute__((ext_vector_type(8))) float v8f;

__device__ __forceinline__ v8f wmma_f32_16x16x4(v2f a, v2f b, v8f c) {
    // 8 args: (neg_a, A, neg_b, B, c_mod, C, reuse_a, reuse_b)
    return __builtin_amdgcn_wmma_f32_16x16x4_f32(
        false, a, false, b, (short)0, c, false, false);
}

#define HDIM 32

// ---------------------------------------------------------------------------
// h = relu(x @ W1 + b1)   — one wave per 16-node tile, 16x32 output per wave.
// ---------------------------------------------------------------------------
__global__ void gmm_lin1_relu(const float* __restrict__ x,
                              const float* __restrict__ W,
                              const float* __restrict__ b,
                              float* __restrict__ h, int nTiles) {
    int wid  = (int)((blockIdx.x * blockDim.x + threadIdx.x) >> 5);
    if (wid >= nTiles) return;                 // wave-uniform: EXEC all-ones
    int lane = threadIdx.x & 31;
    int lo   = lane & 15;                      // col / row-in-half index
    int hi   = lane >> 4;                      // lane half (K split)

    const float* arow = x + (size_t)(wid * 16 + lo) * HDIM;
    v8f c0 = {}, c1 = {};
#pragma unroll
    for (int k0 = 0; k0 < HDIM; k0 += 4) {
        int ka = k0 + hi * 2;
        v2f a;  a.x  = arow[ka];               a.y  = arow[ka + 1];
        v2f b0; b0.x = W[ka * HDIM + lo];      b0.y = W[(ka + 1) * HDIM + lo];
        v2f b1; b1.x = W[ka * HDIM + 16 + lo]; b1.y = W[(ka + 1) * HDIM + 16 + lo];
        c0 = wmma_f32_16x16x4(a, b0, c0);
        c1 = wmma_f32_16x16x4(a, b1, c1);
    }
    float bv0 = b[lo], bv1 = b[16 + lo];
#pragma unroll
    for (int v = 0; v < 8; v++) {
        int row = wid * 16 + v + 8 * hi;
        float r0 = c0[v] + bv0; r0 = r0 > 0.f ? r0 : 0.f;
        float r1 = c1[v] + bv1; r1 = r1 > 0.f ? r1 : 0.f;
        h[(size_t)row * HDIM + lo]      = r0;
        h[(size_t)row * HDIM + 16 + lo] = r1;
    }
}

// ---------------------------------------------------------------------------
// Fused per-layer node GEMMs:
//   hg   = h @ g[layer]
//   base = h @ root[layer] + bias[layer] + fuse[layer] * h
//   agg  = 0   (cleared for the subsequent edge scatter)
// ---------------------------------------------------------------------------
__global__ void gmm_layer_gemm(const float* __restrict__ h,
                               const float* __restrict__ gW,
                               const float* __restrict__ rW,
                               const float* __restrict__ bias,
                               const float* __restrict__ fuseAll, int layer,
                               float* __restrict__ hg,
                               float* __restrict__ base,
                               float* __restrict__ agg, int nTiles) {
    int wid  = (int)((blockIdx.x * blockDim.x + threadIdx.x) >> 5);
    if (wid >= nTiles) return;
    int lane = threadIdx.x & 31;
    int lo   = lane & 15;
    int hi   = lane >> 4;
    float fuse = fuseAll[layer];

    const float* arow = h + (size_t)(wid * 16 + lo) * HDIM;
    v8f cg0 = {}, cg1 = {}, cr0 = {}, cr1 = {};
#pragma unroll
    for (int k0 = 0; k0 < HDIM; k0 += 4) {
        int ka = k0 + hi * 2;
        v2f a;   a.x   = arow[ka];                a.y   = arow[ka + 1];
        v2f bg0; bg0.x = gW[ka * HDIM + lo];      bg0.y = gW[(ka + 1) * HDIM + lo];
        v2f bg1; bg1.x = gW[ka * HDIM + 16 + lo]; bg1.y = gW[(ka + 1) * HDIM + 16 + lo];
        v2f br0; br0.x = rW[ka * HDIM + lo];      br0.y = rW[(ka + 1) * HDIM + lo];
        v2f br1; br1.x = rW[ka * HDIM + 16 + lo]; br1.y = rW[(ka + 1) * HDIM + 16 + lo];
        cg0 = wmma_f32_16x16x4(a, bg0, cg0);
        cg1 = wmma_f32_16x16x4(a, bg1, cg1);
        cr0 = wmma_f32_16x16x4(a, br0, cr0);
        cr1 = wmma_f32_16x16x4(a, br1, cr1);
    }
    float bv0 = bias[lo], bv1 = bias[16 + lo];
#pragma unroll
    for (int v = 0; v < 8; v++) {
        size_t row = (size_t)(wid * 16 + v + 8 * hi);
        size_t i0  = row * HDIM + lo;
        size_t i1  = row * HDIM + 16 + lo;
        float h0 = h[i0], h1 = h[i1];
        hg[i0]   = cg0[v];
        hg[i1]   = cg1[v];
        base[i0] = cr0[v] + bv0 + fuse * h0;
        base[i1] = cr1[v] + bv1 + fuse * h1;
        agg[i0]  = 0.f;
        agg[i1]  = 0.f;
    }
}

// ---------------------------------------------------------------------------
// Degree: zero then float-atomic scatter of ones over dst.
// ---------------------------------------------------------------------------
__global__ void gmm_deg_zero(float* __restrict__ deg, int n) {
    int i = blockIdx.x * blockDim.x + threadIdx.x;
    if (i < n) deg[i] = 0.f;
}

__global__ void gmm_deg_count(const long long* __restrict__ ei,
                              float* __restrict__ deg, int E) {
    int e = blockIdx.x * blockDim.x + threadIdx.x;
    if (e < E) {
        int d = (int)ei[(size_t)E + e];   // row 1 = dst
        __hip_atomic_fetch_add(&deg[d], 1.0f,
                               __ATOMIC_RELAXED, __HIP_MEMORY_SCOPE_AGENT);
    }
}

// ---------------------------------------------------------------------------
// Edge scatter: one wave per edge, lane = feature channel.
//   agg[dst, :] += hg[src, :] * exp(-0.5*(ew-mu)^2/(sigma^2+eps))
// hg row = 128 B = exactly one cacheline (L2-resident gather).
// ---------------------------------------------------------------------------
__global__ void gmm_edge_scatter(const long long* __restrict__ ei,
                                 const float* __restrict__ ew,
                                 const float* __restrict__ muAll,
                                 const float* __restrict__ sigAll, int layer,
                                 const float* __restrict__ hg,
                                 float* __restrict__ agg, int E) {
    int wid = (int)((blockIdx.x * blockDim.x + threadIdx.x) >> 5);
    if (wid >= E) return;
    int lane = threadIdx.x & 31;

    int src = (int)ei[wid];
    int dst = (int)ei[(size_t)E + wid];
    float mu = muAll[layer];
    float sg = sigAll[layer];
    float d  = ew[wid] - mu;
    float gauss = __expf(-0.5f * d * d / (sg * sg + 1e-15f));

    float val = hg[(size_t)src * HDIM + lane] * gauss;
    __hip_atomic_fetch_add(&agg[(size_t)dst * HDIM + lane], val,
                           __ATOMIC_RELAXED, __HIP_MEMORY_SCOPE_AGENT);
}

// ---------------------------------------------------------------------------
// h = agg / max(deg,1) + base     (elementwise over N*32)
// ---------------------------------------------------------------------------
__global__ void gmm_finalize(const float* __restrict__ agg,
                             const float* __restrict__ deg,
                             const float* __restrict__ base,
                             float* __restrict__ h, int n) {
    int i = blockIdx.x * blockDim.x + threadIdx.x;
    if (i < n) {
        float dg = deg[i >> 5];
        dg = dg < 1.f ? 1.f : dg;
        h[i] = agg[i] / dg + base[i];
    }
}

// ---------------------------------------------------------------------------
// out = log_softmax(h @ Wout + bout), C = 2. One thread per node.
// ---------------------------------------------------------------------------
__global__ void gmm_out_head(const float* __restrict__ h,
                             const float* __restrict__ Wout,
                             const float* __restrict__ bout,
                             float* __restrict__ out, int N) {
    int n = blockIdx.x * blockDim.x + threadIdx.x;
    if (n >= N) return;
    float l0 = bout[0], l1 = bout[1];
    const float* hr = h + (size_t)n * HDIM;
#pragma unroll
    for (int j = 0; j < HDIM; j++) {
        float hv = hr[j];
        l0 += hv * Wout[j * 2 + 0];
        l1 += hv * Wout[j * 2 + 1];
    }
    float mx  = fmaxf(l0, l1);
    float lse = mx + __logf(__expf(l0 - mx) + __expf(l1 - mx));
    out[(size_t)n * 2 + 0] = l0 - lse;
    out[(size_t)n * 2 + 1] = l1 - lse;
}

// ---------------------------------------------------------------------------
extern "C" void kernel_launch(void* const* d_in, const int* in_sizes, int n_in,
                              void* d_out, int out_size, void* d_ws, size_t ws_size,
                              hipStream_t stream) {
    const float*     x    = (const float*)d_in[0];
    const long long* ei   = (const long long*)d_in[1];   // int64 [2, E]
    const float*     ew   = (const float*)d_in[2];       // [E, 1]
    const float*     W1   = (const float*)d_in[3];       // [32, 32]
    const float*     b1   = (const float*)d_in[4];       // [32]
    const float*     g    = (const float*)d_in[5];       // [L, 32, 32]
    const float*     mu   = (const float*)d_in[6];       // [L]
    const float*     sig  = (const float*)d_in[7];       // [L]
    const float*     root = (const float*)d_in[8];       // [L, 32, 32]
    const float*     bias = (const float*)d_in[9];       // [L, 32]
    const float*     Wout = (const float*)d_in[10];      // [32, 2]
    const float*     bout = (const float*)d_in[11];      // [2]
    const float*     fuse = (const float*)d_in[12];      // [L]
    float*           out  = (float*)d_out;

    const int N = in_sizes[0] / HDIM;   // x is [N, 32]
    const int E = in_sizes[2];          // edge_weight is [E, 1]
    const int L = in_sizes[6];          // mu is [L]

    // Workspace layout (offsets are 256B-aligned: N*32*4 = 12,800,000 bytes)
    size_t nodeF = (size_t)N * HDIM;
    float* h    = (float*)d_ws;
    float* hg   = h    + nodeF;
    float* base = hg   + nodeF;
    float* agg  = base + nodeF;
    float* deg  = agg  + nodeF;

    const int nTiles     = (N + 15) / 16;
    const int gemmThr    = nTiles * 32;
    const int gemmBlocks = (gemmThr + 255) / 256;

    // 1) h = relu(x @ W1 + b1)
    gmm_lin1_relu<<<gemmBlocks, 256, 0, stream>>>(x, W1, b1, h, nTiles);

    // 2) degree
    gmm_deg_zero<<<(N + 255) / 256, 256, 0, stream>>>(deg, N);
    gmm_deg_count<<<(E + 255) / 256, 256, 0, stream>>>(ei, deg, E);

    // 3) layers
    const long long edgeThr = (long long)E * 32;
    const int edgeBlocks    = (int)((edgeThr + 255) / 256);
    for (int l = 0; l < L; l++) {
        gmm_layer_gemm<<<gemmBlocks, 256, 0, stream>>>(
            h, g + (size_t)l * HDIM * HDIM, root + (size_t)l * HDIM * HDIM,
            bias + (size_t)l * HDIM, fuse, l, hg, base, agg, nTiles);
        gmm_edge_scatter<<<edgeBlocks, 256, 0, stream>>>(
            ei, ew, mu, sig, l, hg, agg, E);
        gmm_finalize<<<((int)nodeF + 255) / 256, 256, 0, stream>>>(
            agg, deg, base, h, (int)nodeF);
    }

    // 4) head + log_softmax
    gmm_out_head<<<(N + 255) / 256, 256, 0, stream>>>(h, Wout, bout, out, N);
}